// CrossAttentionBlock_13872744366803
// MI455X (gfx1250) — compile-verified
//
#include <hip/hip_runtime.h>
#include <hip/hip_bf16.h>

// CrossAttentionBlock for MI455X (gfx1250), full-fp32 pipeline on
// V_WMMA_F32_16X16X4_F32 (problem is bandwidth-bound: ~77 GFLOP vs ~100+ MB
// of HBM traffic, so fp32 tensor ops are free and match the fp32 reference).
//
// Pipeline:
//   1) ln_kernel          : LayerNorm(queries) -> q_ln, LayerNorm(kv) -> kv_ln
//   2) gemm_wmma (x3)     : Q = q_ln @ Wqkv[:, :H],  K/V = kv_ln @ Wqkv slices
//   3) attn_kernel        : flash-attention, 32 q-rows/wave (K,V fragments
//                           shared by two 16-row tiles -> half the L2 traffic),
//                           global_prefetch of the next key chunk
//   4) gemm_wmma          : out = ctx @ W_out + b_out
//
// Workspace layout (floats), total 112 MB (q_ln region reused for ctx):
//   q_ln [2048*1024] | kv_ln [8192*1024] | Q [2048*1024] | K [8192*1024] | V [8192*1024]

typedef __attribute__((ext_vector_type(2))) float v2f;
typedef __attribute__((ext_vector_type(8))) float v8f;

constexpr int kB  = 2;
constexpr int kNQ = 1024;
constexpr int kS  = 4096;
constexpr int kH  = 1024;
constexpr int kNH = 16;
constexpr int kHD = 64;

static __device__ __forceinline__ v8f wmma_f32(v2f a, v2f b, v8f c) {
  // D = A(16x4) * B(4x16) + C(16x16), all fp32.
  return __builtin_amdgcn_wmma_f32_16x16x4_f32(false, a, false, b, (short)0, c,
                                               false, false);
}

// ---------------------------------------------------------------------------
// LayerNorm: one block (256 threads) per row of H=1024.
// ---------------------------------------------------------------------------
__global__ __launch_bounds__(256) void ln_kernel(const float* __restrict__ x,
                                                 const float* __restrict__ gamma,
                                                 const float* __restrict__ beta,
                                                 float* __restrict__ y) {
  const int row = blockIdx.x;
  const float* xr = x + (size_t)row * kH;
  float* yr = y + (size_t)row * kH;

  float v[4];
  float s = 0.f;
#pragma unroll
  for (int i = 0; i < 4; ++i) {
    v[i] = xr[threadIdx.x + 256 * i];
    s += v[i];
  }

  __shared__ float red[256];
  red[threadIdx.x] = s;
  __syncthreads();
  for (int off = 128; off > 0; off >>= 1) {
    if (threadIdx.x < off) red[threadIdx.x] += red[threadIdx.x + off];
    __syncthreads();
  }
  const float mu = red[0] * (1.0f / kH);
  __syncthreads();

  float vs = 0.f;
#pragma unroll
  for (int i = 0; i < 4; ++i) {
    const float d = v[i] - mu;
    vs += d * d;
  }
  red[threadIdx.x] = vs;
  __syncthreads();
  for (int off = 128; off > 0; off >>= 1) {
    if (threadIdx.x < off) red[threadIdx.x] += red[threadIdx.x + off];
    __syncthreads();
  }
  const float rstd = rsqrtf(red[0] * (1.0f / kH) + 1e-6f);

#pragma unroll
  for (int i = 0; i < 4; ++i) {
    const int c = threadIdx.x + 256 * i;
    yr[c] = (v[i] - mu) * rstd * gamma[c] + beta[c];
  }
}

// ---------------------------------------------------------------------------
// Generic fp32 WMMA GEMM: C[M x N] = A[M x K] * W[K x N] + bias[N]
// Grid: (N/64, M/64). Block: 256 threads = 8 waves; each wave owns a 16x32
// sub-tile (two 16x16 WMMA accumulators). A & W staged in LDS (padded).
// Next K-tile is prefetched (global_prefetch_b8) while current one computes.
// ---------------------------------------------------------------------------
__global__ __launch_bounds__(256) void gemm_wmma(const float* __restrict__ A,
                                                 int lda,
                                                 const float* __restrict__ W,
                                                 int ldw,
                                                 const float* __restrict__ bias,
                                                 float* __restrict__ C, int ldc,
                                                 int K) {
  __shared__ float sA[64][34];  // 64 rows x 32 K, pad->34 (conflict-free, 8B ok)
  __shared__ float sW[32][66];  // 32 K x 64 cols, pad->66

  const int tid = threadIdx.x;
  const int wave = tid >> 5;       // 0..7
  const int lane = tid & 31;
  const int h2 = lane >> 4;        // lane half
  const int l16 = lane & 15;
  const int wm = (wave & 3) * 16;  // wave's row offset in tile
  const int wn = (wave >> 2) * 32; // wave's col offset in tile

  const int gm0 = blockIdx.y * 64;
  const int gn0 = blockIdx.x * 64;

  v8f acc0 = {};
  v8f acc1 = {};

  for (int k0 = 0; k0 < K; k0 += 32) {
    // Cooperative stage: A 64x32 (2048 elems) and W 32x64 (2048 elems).
#pragma unroll
    for (int e = 0; e < 8; ++e) {
      const int idx = tid + 256 * e;
      const int r = idx >> 5, c = idx & 31;
      sA[r][c] = A[(size_t)(gm0 + r) * lda + k0 + c];
    }
#pragma unroll
    for (int e = 0; e < 8; ++e) {
      const int idx = tid + 256 * e;
      const int r = idx >> 6, c = idx & 63;
      sW[r][c] = W[(size_t)(k0 + r) * ldw + gn0 + c];
    }

    // Prefetch next K-tile into cache while this tile computes.
    if (k0 + 32 < K) {
      if (tid < 64) {
        // next A tile: 64 rows x 128B
        __builtin_prefetch(&A[(size_t)(gm0 + tid) * lda + k0 + 32], 0, 3);
      } else if (tid < 192) {
        // next 64 W rows (two K-tiles) x 256B (two 128B lines per row)
        const int r = tid - 64;  // 0..127
        __builtin_prefetch(&W[(size_t)(k0 + 32 + (r >> 1)) * ldw + gn0 +
                              (r & 1) * 32], 0, 3);
      }
    }
    __syncthreads();

#pragma unroll
    for (int s = 0; s < 8; ++s) {
      const int kk = 4 * s;
      // A frag: lane holds (K = kk+2h, kk+2h+1) of row (wm + l16).
      const v2f a = *(const v2f*)&sA[wm + l16][kk + 2 * h2];
      // B frags: lane holds (B[kk+2h][n], B[kk+2h+1][n]) with n = col.
      v2f b0, b1;
      b0.x = sW[kk + 2 * h2][wn + l16];
      b0.y = sW[kk + 2 * h2 + 1][wn + l16];
      b1.x = sW[kk + 2 * h2][wn + 16 + l16];
      b1.y = sW[kk + 2 * h2 + 1][wn + 16 + l16];
      acc0 = wmma_f32(a, b0, acc0);
      acc1 = wmma_f32(a, b1, acc1);
    }
    __syncthreads();
  }

  // C layout: vgpr i -> row (i + 8*half), col = l16 (+16 for second tile).
#pragma unroll
  for (int i = 0; i < 8; ++i) {
    const int row = gm0 + wm + 8 * h2 + i;
    const int c0 = gn0 + wn + l16;
    const int c1 = c0 + 16;
    C[(size_t)row * ldc + c0] = acc0[i] + bias[c0];
    C[(size_t)row * ldc + c1] = acc1[i] + bias[c1];
  }
}

// ---------------------------------------------------------------------------
// Flash attention, fp32 WMMA. One wave (32 threads) per (b, head, 32 q-rows).
// Two 16-row q-subtiles share every K/V B-fragment (halves L2 traffic).
// Streams keys in chunks of 16 with online softmax; P converted C->A layout
// via a small LDS round-trip; next chunk's K/V rows prefetched.
// ---------------------------------------------------------------------------
__global__ __launch_bounds__(32) void attn_kernel(
    const float* __restrict__ Q,  // [B*NQ, H]
    const float* __restrict__ K,  // [B*S,  H]
    const float* __restrict__ V,  // [B*S,  H]
    const unsigned char* __restrict__ mask,  // [B, S] (jax bool, 1 byte)
    float* __restrict__ ctx) {    // [B*NQ, H]
  const int qt = blockIdx.x;   // q tile (32 rows)
  const int hh = blockIdx.y;   // head
  const int b  = blockIdx.z;

  const int lane = threadIdx.x;
  const int h2 = lane >> 4;
  const int l16 = lane & 15;

  const float* Qp = Q + ((size_t)b * kNQ + qt * 32) * kH + hh * kHD;
  const float* Kp = K + (size_t)b * kS * kH + hh * kHD;
  const float* Vp = V + (size_t)b * kS * kH + hh * kHD;
  const unsigned char* mb = mask + (size_t)b * kS;

  // Preload Q as A-fragments for both 16-row subtiles, pre-scaled 1/sqrt(HD).
  v2f aq[2][16];
#pragma unroll
  for (int u = 0; u < 2; ++u)
#pragma unroll
    for (int s = 0; s < 16; ++s) {
      aq[u][s] =
          *(const v2f*)&Qp[(size_t)(u * 16 + l16) * kH + 4 * s + 2 * h2];
      aq[u][s] *= 0.125f;  // 1/sqrt(64)
    }

  v8f o[2][4];
  float m_i[2][8], l_i[2][8];
#pragma unroll
  for (int u = 0; u < 2; ++u) {
#pragma unroll
    for (int t = 0; t < 4; ++t) o[u][t] = (v8f){};
#pragma unroll
    for (int i = 0; i < 8; ++i) {
      m_i[u][i] = -3.4e38f;
      l_i[u][i] = 0.f;
    }
  }

  __shared__ float sP[32][18];  // P staging (pad 18: even stride, no conflicts)

  for (int kb = 0; kb < kS; kb += 16) {
    // Prefetch next chunk's K and V rows (each row = 256B = 2 cachelines;
    // the two lane-halves cover both halves of the row).
    if (kb + 16 < kS) {
      const char* nk = (const char*)(Kp + (size_t)(kb + 16 + l16) * kH);
      const char* nv = (const char*)(Vp + (size_t)(kb + 16 + l16) * kH);
      __builtin_prefetch(nk + h2 * 128, 0, 3);
      __builtin_prefetch(nv + h2 * 128, 0, 3);
    }

    // --- scores: two 16x16 tiles share each K B-fragment.
    v8f sc0 = {}, sc1 = {};
    const float* krow = Kp + (size_t)(kb + l16) * kH;
#pragma unroll
    for (int s = 0; s < 16; ++s) {
      const v2f bk = *(const v2f*)&krow[4 * s + 2 * h2];
      sc0 = wmma_f32(aq[0][s], bk, sc0);
      sc1 = wmma_f32(aq[1][s], bk, sc1);
    }

    const float madd = mb[kb + l16] ? 0.0f : -3.0e38f;

    // --- online softmax per row r = i + 8*h2 (16 cols live in this lane-half)
    float alpha[2][8];
#pragma unroll
    for (int u = 0; u < 2; ++u) {
      v8f& sc = (u == 0) ? sc0 : sc1;
#pragma unroll
      for (int i = 0; i < 8; ++i) {
        float sv = sc[i] + madd;
        float rmax = sv;
#pragma unroll
        for (int off = 1; off < 16; off <<= 1)
          rmax = fmaxf(rmax, __shfl_xor(rmax, off, 32));
        const float mnew = fmaxf(m_i[u][i], rmax);
        const float p = __expf(sv - mnew);
        const float a_old = __expf(m_i[u][i] - mnew);
        float rsum = p;
#pragma unroll
        for (int off = 1; off < 16; off <<= 1)
          rsum += __shfl_xor(rsum, off, 32);
        l_i[u][i] = l_i[u][i] * a_old + rsum;
        m_i[u][i] = mnew;
        alpha[u][i] = a_old;
        sc[i] = p;
      }
      // Rescale accumulated context (row-matched: vgpr i -> row i+8h).
#pragma unroll
      for (int t = 0; t < 4; ++t)
#pragma unroll
        for (int i = 0; i < 8; ++i) o[u][t][i] *= alpha[u][i];
      // P: C-layout -> LDS (subtile u occupies rows u*16 .. u*16+15).
#pragma unroll
      for (int i = 0; i < 8; ++i) sP[u * 16 + 8 * h2 + i][l16] = sc[i];
    }
    __syncthreads();

    // o(32x64) += P(32x16) x V(16x64): V B-fragments shared by both subtiles.
#pragma unroll
    for (int s2 = 0; s2 < 4; ++s2) {
      const v2f ap0 = *(const v2f*)&sP[l16][4 * s2 + 2 * h2];
      const v2f ap1 = *(const v2f*)&sP[16 + l16][4 * s2 + 2 * h2];
      const float* vr = Vp + (size_t)(kb + 4 * s2 + 2 * h2) * kH;
#pragma unroll
      for (int t = 0; t < 4; ++t) {
        v2f bv;
        bv.x = vr[t * 16 + l16];
        bv.y = vr[kH + t * 16 + l16];
        o[0][t] = wmma_f32(ap0, bv, o[0][t]);
        o[1][t] = wmma_f32(ap1, bv, o[1][t]);
      }
    }
    __syncthreads();
  }

  // Epilogue: divide by softmax denominator, store context.
#pragma unroll
  for (int u = 0; u < 2; ++u)
#pragma unroll
    for (int t = 0; t < 4; ++t)
#pragma unroll
      for (int i = 0; i < 8; ++i) {
        const size_t row = (size_t)b * kNQ + qt * 32 + u * 16 + 8 * h2 + i;
        ctx[row * kH + hh * kHD + t * 16 + l16] = o[u][t][i] / l_i[u][i];
      }
}

// ---------------------------------------------------------------------------
extern "C" void kernel_launch(void* const* d_in, const int* in_sizes, int n_in,
                              void* d_out, int out_size, void* d_ws,
                              size_t ws_size, hipStream_t stream) {
  (void)in_sizes; (void)n_in; (void)out_size; (void)ws_size;

  const float* queries     = (const float*)d_in[0];  // [B, NQ, H]
  const float* keys_values = (const float*)d_in[1];  // [B, S, H]
  const unsigned char* am  = (const unsigned char*)d_in[2];  // [B, S] bool
  const float* W_qkv       = (const float*)d_in[3];  // [H, 3H]
  const float* b_qkv       = (const float*)d_in[4];  // [3H]
  const float* W_out       = (const float*)d_in[5];  // [H, H]
  const float* b_out       = (const float*)d_in[6];  // [H]
  const float* gamma       = (const float*)d_in[7];  // [H]
  const float* beta        = (const float*)d_in[8];  // [H]
  float* out = (float*)d_out;

  float* ws = (float*)d_ws;
  float* q_ln  = ws;                                   // 2048*1024
  float* kv_ln = q_ln  + (size_t)kB * kNQ * kH;        // 8192*1024
  float* Qb    = kv_ln + (size_t)kB * kS * kH;         // 2048*1024
  float* Kb    = Qb    + (size_t)kB * kNQ * kH;        // 8192*1024
  float* Vb    = Kb    + (size_t)kB * kS * kH;         // 8192*1024
  float* ctx   = q_ln;  // q_ln dead after QKV projections; reuse (112 MB total)

  // 1) LayerNorms
  ln_kernel<<<kB * kNQ, 256, 0, stream>>>(queries, gamma, beta, q_ln);
  ln_kernel<<<kB * kS, 256, 0, stream>>>(keys_values, gamma, beta, kv_ln);

  // 2) QKV projections (column slices of W_qkv; row-major ld = 3H)
  dim3 gq(kH / 64, (kB * kNQ) / 64);
  dim3 gkv(kH / 64, (kB * kS) / 64);
  gemm_wmma<<<gq, 256, 0, stream>>>(q_ln, kH, W_qkv + 0 * kH, 3 * kH,
                                    b_qkv + 0 * kH, Qb, kH, kH);
  gemm_wmma<<<gkv, 256, 0, stream>>>(kv_ln, kH, W_qkv + 1 * kH, 3 * kH,
                                     b_qkv + 1 * kH, Kb, kH, kH);
  gemm_wmma<<<gkv, 256, 0, stream>>>(kv_ln, kH, W_qkv + 2 * kH, 3 * kH,
                                     b_qkv + 2 * kH, Vb, kH, kH);

  // 3) Flash attention: grid (32-row q-tiles, heads, batch), 1 wave per block
  dim3 ga(kNQ / 32, kNH, kB);
  attn_kernel<<<ga, 32, 0, stream>>>(Qb, Kb, Vb, am, ctx);

  // 4) Output projection -> d_out
  gemm_wmma<<<gq, 256, 0, stream>>>(ctx, kH, W_out, kH, b_out, out, kH, kH);
}